// GlobalAttentionPooling_26542897889306
// MI455X (gfx1250) — compile-verified
//
#include <hip/hip_runtime.h>
#include <hip/hip_bf16.h>

typedef __attribute__((ext_vector_type(2))) float v2f;
typedef __attribute__((ext_vector_type(4))) float v4f;
typedef __attribute__((ext_vector_type(8))) float v8f;

#define DCH 256
#define GATE_CAP 4096
#define NEG_HUGE -3.0e38f

// ---------------------------------------------------------------------------
// Kernel 1: segment boundaries via binary search (segment_ids sorted).
// bounds[b] = first index i with seg[i] >= b ; bounds[B] = N.
// ---------------------------------------------------------------------------
__global__ void seg_bounds_kernel(const int* __restrict__ seg, int N, int B,
                                  int* __restrict__ bounds) {
    int b = blockIdx.x * blockDim.x + threadIdx.x;
    if (b > B) return;
    if (b == B) { bounds[B] = N; return; }
    int lo = 0, hi = N;
    while (lo < hi) {
        int mid = (lo + hi) >> 1;
        if (seg[mid] < b) lo = mid + 1; else hi = mid;
    }
    bounds[b] = lo;
}

// ---------------------------------------------------------------------------
// Kernel 2: fused gate GEMV (WMMA) + segment softmax + weighted readout.
// One block (256 threads = 8 wave32) per segment.
//
// Pass 1 WMMA trick: A[m,k] = W[k0+k] for all m (replicated W chunk),
// B[k,n] = feat[node_n][k0+k]. Then C[m][n] = gate[node_n] for every m;
// per the 16x16 f32 C layout, c[0] gives each lane gate[node_(lane&15)].
//
// Pass 2: thread t = r*64+q owns channels 4q..4q+3 of row-slot r; one b128
// non-temporal load + one broadcast LDS read + 4 FMAs per 4 rows.
// ---------------------------------------------------------------------------
__global__ __launch_bounds__(256)
void gap_fused_kernel(const float* __restrict__ feat,
                      const float* __restrict__ Wg,
                      const float* __restrict__ bgate,
                      const int*   __restrict__ bounds,
                      float*       __restrict__ out) {
    __shared__ float s_gate[GATE_CAP];
    __shared__ float s_W[DCH];
    __shared__ float s_m[8];
    __shared__ float s_s[8];
    __shared__ v4f   s_red[256];

    const int b    = blockIdx.x;
    const int tid  = threadIdx.x;
    const int wave = tid >> 5;
    const int lane = tid & 31;
    const int j    = lane & 15;   // node slot within 16-node WMMA tile
    const int hf   = lane >> 4;   // wave half (K split for A/B fragments)

    const int start = bounds[b];
    const int end   = bounds[b + 1];
    const int cnt   = end - start;

    s_W[tid] = Wg[tid];           // blockDim == DCH == 256
    __syncthreads();

    const float bias = bgate[0];

    // ---------------- Pass 1: WMMA gates + per-wave online softmax ----------
    float m = NEG_HUGE, s = 0.0f;

    for (int base = start + wave * 16; base < end; base += 8 * 16) {
        const int node    = base + j;
        const bool valid  = node < end;
        const int rowNode = valid ? node : (end - 1);      // clamp (discarded)
        const float* row  = feat + (size_t)rowNode * DCH + 2 * hf;

        v8f c = {};
        #pragma unroll
        for (int k0 = 0; k0 < DCH; k0 += 4) {
            // B fragment: rows K=0/2 in vgpr0, K=1/3 in vgpr1 (per wave half)
            v2f bf = *(const v2f*)(row + k0);
            // A fragment: replicated W chunk
            v2f af;
            af.x = s_W[k0 + 2 * hf];
            af.y = s_W[k0 + 2 * hf + 1];
            c = __builtin_amdgcn_wmma_f32_16x16x4_f32(
                    false, af, false, bf, (short)0, c, false, false);
        }
        const float g = c[0] + bias;   // gate[node_(lane&15)] in every lane

        if (valid && hf == 0) {
            const int idx = node - start;
            if (idx < GATE_CAP) s_gate[idx] = g;
        }
        const float gv = valid ? g : NEG_HUGE;
        const float nm = fmaxf(m, gv);
        s = s * __expf(m - nm) + (valid ? __expf(gv - nm) : 0.0f);
        m = nm;
    }

    // Reduce (m,s) across the 16-lane group (halves hold duplicates).
    #pragma unroll
    for (int off = 1; off < 16; off <<= 1) {
        const float mo = __shfl_xor(m, off, 32);
        const float so = __shfl_xor(s, off, 32);
        const float nm = fmaxf(m, mo);
        s = s * __expf(m - nm) + so * __expf(mo - nm);
        m = nm;
    }
    if (lane == 0) { s_m[wave] = m; s_s[wave] = s; }
    __syncthreads();

    // Block-level combine (every thread computes the same result).
    float M = NEG_HUGE;
    #pragma unroll
    for (int w = 0; w < 8; ++w) M = fmaxf(M, s_m[w]);
    float S = 0.0f;
    #pragma unroll
    for (int w = 0; w < 8; ++w) S += s_s[w] * __expf(s_m[w] - M);
    const float invS = (S > 0.0f) ? (1.0f / S) : 0.0f;
    __syncthreads();

    // Convert cached gates -> alpha in LDS.
    const int cc = (cnt < GATE_CAP) ? cnt : GATE_CAP;
    for (int i = tid; i < cc; i += 256)
        s_gate[i] = __expf(s_gate[i] - M) * invS;
    __syncthreads();

    // ---------------- Pass 2: weighted readout, wide loads ------------------
    const int q = tid & 63;       // channel group: owns channels 4q..4q+3
    const int r = tid >> 6;       // row slot 0..3 (wave-uniform)
    const float* fb = feat + (size_t)start * DCH + 4 * q;

    v4f acc = {0.0f, 0.0f, 0.0f, 0.0f};

    int i = 0;
    for (; i + 8 <= cc; i += 8) {                    // 8 rows per iteration
        const float a0 = s_gate[i + r];
        const float a1 = s_gate[i + 4 + r];
        const v4f f0 = __builtin_nontemporal_load(
                           (const v4f*)(fb + (size_t)(i + r) * DCH));
        const v4f f1 = __builtin_nontemporal_load(
                           (const v4f*)(fb + (size_t)(i + 4 + r) * DCH));
        acc += a0 * f0;
        acc += a1 * f1;
    }
    for (; i < cc; i += 4) {                          // tail (wave-uniform r)
        if (i + r < cc) {
            const float a = s_gate[i + r];
            const v4f f = __builtin_nontemporal_load(
                              (const v4f*)(fb + (size_t)(i + r) * DCH));
            acc += a * f;
        }
    }

    // Overflow path (cnt > GATE_CAP): statistically never taken, kept for
    // correctness. Only the r==0 group accumulates (avoids double count).
    for (int o = cc; o < cnt; ++o) {
        const float* row = feat + (size_t)(start + o) * DCH;
        float g = bias;
        for (int k = 0; k < DCH; ++k) g += row[k] * s_W[k];
        const float a = __expf(g - M) * invS;
        if (r == 0) {
            const v4f f = *(const v4f*)(row + 4 * q);
            acc += a * f;
        }
    }

    // Reduce the 4 row-slot partials per channel group and store b128.
    s_red[tid] = acc;
    __syncthreads();
    if (tid < 64) {
        const v4f p0 = s_red[tid];
        const v4f p1 = s_red[tid + 64];
        const v4f p2 = s_red[tid + 128];
        const v4f p3 = s_red[tid + 192];
        const v4f o  = (p0 + p1) + (p2 + p3);
        *(v4f*)(out + (size_t)b * DCH + 4 * tid) = o;
    }
}

// ---------------------------------------------------------------------------
extern "C" void kernel_launch(void* const* d_in, const int* in_sizes, int n_in,
                              void* d_out, int out_size, void* d_ws, size_t ws_size,
                              hipStream_t stream) {
    const float* feat  = (const float*)d_in[0];
    const float* Wg    = (const float*)d_in[1];
    const float* bgate = (const float*)d_in[2];
    const int*   seg   = (const int*)d_in[3];

    const int N = in_sizes[3];          // number of nodes
    const int B = out_size / DCH;       // number of segments (1024)

    int* bounds = (int*)d_ws;           // (B+1) ints

    const int T = 256;
    const int blocks = (B + 1 + T - 1) / T;
    seg_bounds_kernel<<<blocks, T, 0, stream>>>(seg, N, B, bounds);

    gap_fused_kernel<<<B, T, 0, stream>>>(feat, Wg, bgate, bounds, (float*)d_out);
}